// PhysicsInformedNN_42992622633175
// MI455X (gfx1250) — compile-verified
//
#include <hip/hip_runtime.h>
#include <math.h>

typedef __attribute__((ext_vector_type(2))) float v2f;
typedef __attribute__((ext_vector_type(8))) float v8f;

#define NPTS 16          // points per workgroup
#define NJET 13          // jet components per neuron
#define COLS (NPTS * NJET)   // 208 GEMM columns
#define CSTR 132             // padded LDS column stride (floats) -> conflict-free
#define HBUF (COLS * CSTR)   // floats per ping-pong buffer

// jet component ordering
#define JV   0
#define JX   1
#define JY   2
#define JT   3
#define JXX  4
#define JXY  5
#define JYY  6
#define JXT  7
#define JYT  8
#define JXXX 9
#define JXXY 10
#define JXYY 11
#define JYYY 12

// Prefer the native CDNA5 v_tanh_f32 transcendental (co-executes with the
// WMMA pipe); fall back to OCML tanhf if the builtin isn't declared.
#if __has_builtin(__builtin_amdgcn_tanhf)
__device__ __forceinline__ float fast_tanh(float x) { return __builtin_amdgcn_tanhf(x); }
#elif __has_builtin(__builtin_amdgcn_tanh_f32)
__device__ __forceinline__ float fast_tanh(float x) { return __builtin_amdgcn_tanh_f32(x); }
#else
__device__ __forceinline__ float fast_tanh(float x) { return tanhf(x); }
#endif

__global__ __launch_bounds__(256)
void pinn_jet_kernel(const float* __restrict__ W0, const float* __restrict__ b0,
                     const float* __restrict__ W1, const float* __restrict__ b1,
                     const float* __restrict__ W2, const float* __restrict__ b2,
                     const float* __restrict__ W3, const float* __restrict__ b3,
                     const float* __restrict__ W4, const float* __restrict__ b4,
                     const float* __restrict__ W5, const float* __restrict__ b5,
                     const float* __restrict__ W6, const float* __restrict__ b6,
                     const float* __restrict__ W7, const float* __restrict__ b7,
                     const float* __restrict__ W8, const float* __restrict__ b8,
                     const float* __restrict__ xin, const float* __restrict__ yin,
                     const float* __restrict__ tin,
                     const float* __restrict__ lam1p, const float* __restrict__ lam2p,
                     float* __restrict__ out, int N)
{
    extern __shared__ float smem[];
    float* buf0 = smem;          // ping
    float* buf1 = smem + HBUF;   // pong

    const int tid  = threadIdx.x;
    const int lane = tid & 31;
    const int wave = tid >> 5;
    const int lh   = lane & 15;   // low-half lane id (M/N index)
    const int hh   = lane >> 4;   // which half of the wave (K-split)
    const int p0   = blockIdx.x * NPTS;

    // ---------------- layer 0: 3 -> 128, scalar jet seed ----------------
    // input jets: value=(x,y,t), d/dx=(1,0,0), d/dy=(0,1,0), d/dt=(0,0,1),
    // all higher derivatives zero -> pre-act a_i = W0[:,i], a_ij = a_ijk = 0.
#pragma unroll
    for (int it = 0; it < 8; ++it) {
        int idx = it * 256 + tid;       // 0..2047 = 16 pts x 128 neurons
        int q = idx >> 7;               // point in tile
        int n = idx & 127;              // neuron
        float xz = xin[p0 + q], yz = yin[p0 + q], tz = tin[p0 + q];
        float wx = W0[n * 3 + 0], wy = W0[n * 3 + 1], wt = W0[n * 3 + 2];
        float a  = wx * xz + wy * yz + wt * tz + b0[n];
        float s  = fast_tanh(a);
        float d1 = 1.f - s * s;
        float d2 = -2.f * s * d1;
        float d3 = d1 * (6.f * s * s - 2.f);
        float* c = buf0 + n;  // column-major: col*CSTR + neuron
        c[(JV   * NPTS + q) * CSTR] = s;
        c[(JX   * NPTS + q) * CSTR] = d1 * wx;
        c[(JY   * NPTS + q) * CSTR] = d1 * wy;
        c[(JT   * NPTS + q) * CSTR] = d1 * wt;
        c[(JXX  * NPTS + q) * CSTR] = d2 * wx * wx;
        c[(JXY  * NPTS + q) * CSTR] = d2 * wx * wy;
        c[(JYY  * NPTS + q) * CSTR] = d2 * wy * wy;
        c[(JXT  * NPTS + q) * CSTR] = d2 * wx * wt;
        c[(JYT  * NPTS + q) * CSTR] = d2 * wy * wt;
        c[(JXXX * NPTS + q) * CSTR] = d3 * wx * wx * wx;
        c[(JXXY * NPTS + q) * CSTR] = d3 * wx * wx * wy;
        c[(JXYY * NPTS + q) * CSTR] = d3 * wx * wy * wy;
        c[(JYYY * NPTS + q) * CSTR] = d3 * wy * wy * wy;
    }
    __syncthreads();

    // ---------------- layers 1..7: 128x128 WMMA GEMM + tanh jet ----------------
    const float* Ws[8] = {nullptr, W1, W2, W3, W4, W5, W6, W7};
    const float* bs[8] = {nullptr, b1, b2, b3, b4, b5, b6, b7};
    const int m0 = wave * 16;           // this wave's output-neuron rows

#pragma clang loop unroll(disable)
    for (int L = 1; L <= 7; ++L) {
        const float* W   = Ws[L];
        const float* b   = bs[L];
        const float* src = (L & 1) ? buf0 : buf1;
        float*       dst = (L & 1) ? buf1 : buf0;

        // A fragments: W rows m0..m0+15, all K. 16x4 f32 A layout:
        // lane = (m - m0) + 16*(k/2), vgpr = k%2  -> lane lh,hh holds k = 4kk+2hh, +1
        v2f af[32];
#pragma unroll
        for (int kk = 0; kk < 32; ++kk)
            af[kk] = *(const v2f*)(W + (m0 + lh) * 128 + 4 * kk + 2 * hh);

        float bias[8];
#pragma unroll
        for (int r = 0; r < 8; ++r)
            bias[r] = b[m0 + hh * 8 + r];

        // 13 accumulator tiles, one per jet component (same 16 points each)
        v8f acc[NJET];
#pragma unroll
        for (int j = 0; j < NJET; ++j) {
#pragma unroll
            for (int e = 0; e < 8; ++e) acc[j][e] = 0.f;
#pragma unroll
            for (int kk = 0; kk < 32; ++kk) {
                // B 4x16 f32: lane lh = column, halves split K -> one ds_load_b64
                v2f bf = *(const v2f*)(src + (j * NPTS + lh) * CSTR + 4 * kk + 2 * hh);
                acc[j] = __builtin_amdgcn_wmma_f32_16x16x4_f32(
                    false, af[kk], false, bf, (short)0, acc[j], false, false);
            }
        }

        // tanh jet composition, pure register math (C/D element (r,l):
        // neuron = m0 + r + 8*hh, point = lh -- identical across all 13 tiles)
#pragma unroll
        for (int e = 0; e < 8; ++e) {
            float a0   = acc[JV][e] + bias[e];
            float ax   = acc[JX][e],   ay   = acc[JY][e],   at   = acc[JT][e];
            float axx  = acc[JXX][e],  axy  = acc[JXY][e],  ayy  = acc[JYY][e];
            float axt  = acc[JXT][e],  ayt  = acc[JYT][e];
            float axxx = acc[JXXX][e], axxy = acc[JXXY][e];
            float axyy = acc[JXYY][e], ayyy = acc[JYYY][e];
            float s  = fast_tanh(a0);
            float d1 = 1.f - s * s;
            float d2 = -2.f * s * d1;
            float d3 = d1 * (6.f * s * s - 2.f);
            acc[JV][e]   = s;
            acc[JX][e]   = d1 * ax;
            acc[JY][e]   = d1 * ay;
            acc[JT][e]   = d1 * at;
            acc[JXX][e]  = d2 * ax * ax + d1 * axx;
            acc[JXY][e]  = d2 * ax * ay + d1 * axy;
            acc[JYY][e]  = d2 * ay * ay + d1 * ayy;
            acc[JXT][e]  = d2 * ax * at + d1 * axt;
            acc[JYT][e]  = d2 * ay * at + d1 * ayt;
            acc[JXXX][e] = d3 * ax * ax * ax + 3.f * d2 * axx * ax + d1 * axxx;
            acc[JXXY][e] = d3 * ax * ax * ay + d2 * (axx * ay + 2.f * axy * ax) + d1 * axxy;
            acc[JXYY][e] = d3 * ax * ay * ay + d2 * (ayy * ax + 2.f * axy * ay) + d1 * axyy;
            acc[JYYY][e] = d3 * ay * ay * ay + 3.f * d2 * ayy * ay + d1 * ayyy;
        }

        // store: 8 contiguous neurons per (jet, point) column
#pragma unroll
        for (int j = 0; j < NJET; ++j) {
            float* c = dst + (j * NPTS + lh) * CSTR + m0 + hh * 8;
#pragma unroll
            for (int e = 0; e < 8; ++e) c[e] = acc[j][e];
        }
        __syncthreads();
    }

    // ---------------- layer 8: 128 -> 2, scalar dots ----------------
    const float* hsrc = buf1;   // 7 layers -> final hidden jets in buf1
    float* fin = buf0;          // reuse ping buffer for 2 x 13 x 16 results
#pragma clang loop unroll(disable)
    for (int idx = tid; idx < 2 * NJET * NPTS; idx += 256) {
        int j   = idx >> 5;       // jet component
        int rem = idx & 31;
        int o   = rem >> 4;       // output row: 0=psi, 1=p
        int q   = rem & 15;       // point
        const float* c = hsrc + (j * NPTS + q) * CSTR;
        const float* w = W8 + o * 128;
        float d = 0.f;
#pragma unroll 8
        for (int k = 0; k < 128; ++k) d += w[k] * c[k];
        if (j == JV) d += b8[o];
        fin[(o * NJET + j) * NPTS + q] = d;
    }
    __syncthreads();

    // ---------------- combine into u, v, p, f_u, f_v ----------------
    if (tid < NPTS) {
        int q  = tid;
        int pi = p0 + q;
        float P[NJET];
#pragma unroll
        for (int j = 0; j < NJET; ++j) P[j] = fin[j * NPTS + q];
        float pr = fin[(NJET + JV) * NPTS + q];
        float px = fin[(NJET + JX) * NPTS + q];
        float py = fin[(NJET + JY) * NPTS + q];
        float l1 = lam1p[0], l2 = lam2p[0];

        float u    =  P[JY];
        float v    = -P[JX];
        float u_x  =  P[JXY], u_y =  P[JYY], u_t =  P[JYT];
        float v_x  = -P[JXX], v_y = -P[JXY], v_t = -P[JXT];
        float u_xx =  P[JXXY], u_yy =  P[JYYY];
        float v_xx = -P[JXXX], v_yy = -P[JXYY];

        float f_u = u_t + l1 * (u * u_x + v * u_y) + px - l2 * (u_xx + u_yy);
        float f_v = v_t + l1 * (u * v_x + v * v_y) + py - l2 * (v_xx + v_yy);

        out[pi]         = u;
        out[N + pi]     = v;
        out[2 * N + pi] = pr;
        out[3 * N + pi] = f_u;
        out[4 * N + pi] = f_v;
    }
}

extern "C" void kernel_launch(void* const* d_in, const int* in_sizes, int n_in,
                              void* d_out, int out_size, void* d_ws, size_t ws_size,
                              hipStream_t stream) {
    (void)n_in; (void)out_size; (void)d_ws; (void)ws_size;
    const float* W[9];
    const float* B[9];
    for (int i = 0; i < 9; ++i) {
        W[i] = (const float*)d_in[2 * i];
        B[i] = (const float*)d_in[2 * i + 1];
    }
    const float* x  = (const float*)d_in[18];
    const float* y  = (const float*)d_in[19];
    const float* t  = (const float*)d_in[20];
    const float* l1 = (const float*)d_in[21];
    const float* l2 = (const float*)d_in[22];
    float* out = (float*)d_out;
    int N = in_sizes[18];
    int nblk = N / NPTS;
    size_t shmem = (size_t)(2 * HBUF) * sizeof(float);  // ~214.5 KB of 320 KB LDS/WGP

    pinn_jet_kernel<<<nblk, 256, shmem, stream>>>(
        W[0], B[0], W[1], B[1], W[2], B[2], W[3], B[3], W[4], B[4],
        W[5], B[5], W[6], B[6], W[7], B[7], W[8], B[8],
        x, y, t, l1, l2, out, N);
}